// NodeWeightLayer_75952201663109
// MI455X (gfx1250) — compile-verified
//
#include <hip/hip_runtime.h>
#include <math.h>

typedef float v2f __attribute__((ext_vector_type(2)));
typedef float v8f __attribute__((ext_vector_type(8)));

#define BB 8
#define NN 32768
#define DD 128
#define HH 64
#define KK 32

// ---------------- kernel 0: weight-norm prep (one tiny block) ----------------
// K-pair interleaved layout for WMMA B-fragments:
//   w1P[(k>>1)*128 + c*2 + (k&1)] = g1[c]*v1[c][k]/||v1[c]||
// so a B-frag (K=ka, K=ka+1 for hidden col c) is one aligned 8-byte LDS load.
__global__ void k_prep(const float* __restrict__ v1, const float* __restrict__ g1,
                       const float* __restrict__ v2, const float* __restrict__ g2,
                       float* __restrict__ w1P, float* __restrict__ w2n) {
  int c = threadIdx.x;                       // 0..63, one hidden unit per thread
  float ss = 0.f;
  for (int k = 0; k < DD; ++k) { float v = v1[c * DD + k]; ss = fmaf(v, v, ss); }
  float sc = g1[c] / sqrtf(ss);
  for (int k = 0; k < DD; ++k)
    w1P[(k >> 1) * (2 * HH) + c * 2 + (k & 1)] = v1[c * DD + k] * sc;

  __shared__ float red[HH];
  float vv = v2[c];
  red[c] = vv * vv;
  __syncthreads();
  for (int s = HH / 2; s > 0; s >>= 1) {
    if (c < s) red[c] += red[c + s];
    __syncthreads();
  }
  w2n[c] = g2[0] * vv / sqrtf(red[0]);
}

// ---------------- kernel 1: fused MLP -> per-node logit via FP32 WMMA --------
// Each wave: 16 node rows. Four 16x16 f32 C tiles (hidden cols 0..63),
// K=128 accumulated with v_wmma_f32_16x16x4_f32.
__global__ __launch_bounds__(256) void k_mlp(const float* __restrict__ x,
                                             const float* __restrict__ w1P,
                                             const float* __restrict__ w2n,
                                             const float* __restrict__ b1,
                                             float* __restrict__ logits) {
  __shared__ float w[DD * HH];               // 32 KB: pair-interleaved w1
  const int t = threadIdx.x;
  for (int i = t; i < DD * HH; i += 256) w[i] = w1P[i];
  __syncthreads();

  const int lane  = t & 31;
  const int wave  = t >> 5;
  const int laneM = lane & 15;               // M (A) / N (B,C) index
  const int half  = lane >> 4;               // 0: K+0/K+1, 1: K+2/K+3
  const long rowBase = (long)blockIdx.x * 128 + wave * 16;
  const float* xr = x + (rowBase + laneM) * DD;
  // LDS base for this lane: pair row (half) within each k0 group, column laneM
  const float* wl = &w[half * (2 * HH) + laneM * 2];

  v8f acc[4] = {};                           // cols [0,16), [16,32), [32,48), [48,64)
  for (int k0 = 0; k0 < DD; k0 += 4) {
    const int ka = k0 + half * 2;
    v2f a = *reinterpret_cast<const v2f*>(xr + ka);          // A frag: K=ka, ka+1
    const float* wk = wl + (k0 >> 1) * (2 * HH);             // pair row for this step
#pragma unroll
    for (int j = 0; j < 4; ++j) {
      v2f b = *reinterpret_cast<const v2f*>(wk + j * 32);    // B frag: one ds_load_b64
      acc[j] = __builtin_amdgcn_wmma_f32_16x16x4_f32(
          false, a, false, b, (short)0, acc[j], false, false);
    }
  }

  float bia[4], w2c[4];
#pragma unroll
  for (int j = 0; j < 4; ++j) { bia[j] = b1[j * 16 + laneM]; w2c[j] = w2n[j * 16 + laneM]; }

  // epilogue: relu(h + b1) . w2 , reduce the 16 hidden cols held across laneM
#pragma unroll
  for (int r = 0; r < 8; ++r) {              // C VGPR r: M = r + half*8
    float s = 0.f;
#pragma unroll
    for (int j = 0; j < 4; ++j) {
      float h = acc[j][r] + bia[j];
      s = fmaf(fmaxf(h, 0.f), w2c[j], s);
    }
    s += __shfl_xor(s, 1, 32);
    s += __shfl_xor(s, 2, 32);
    s += __shfl_xor(s, 4, 32);
    s += __shfl_xor(s, 8, 32);
    if (laneM == 0) logits[rowBase + r + half * 8] = s;
  }
}

// ---------------- kernel 2: per-batch softmax stats (max, sum) ---------------
__global__ __launch_bounds__(512) void k_stats(const float* __restrict__ logits,
                                               float* __restrict__ maxb,
                                               float* __restrict__ sumb) {
  __shared__ float red[512];
  const int b = blockIdx.x, t = threadIdx.x;
  const float* L = logits + (long)b * NN;
  float m = -INFINITY;
  for (int i = t; i < NN; i += 512) m = fmaxf(m, L[i]);
  red[t] = m; __syncthreads();
  for (int s = 256; s > 0; s >>= 1) { if (t < s) red[t] = fmaxf(red[t], red[t + s]); __syncthreads(); }
  const float M = red[0]; __syncthreads();
  float sm = 0.f;
  for (int i = t; i < NN; i += 512) sm += __expf(L[i] - M);
  red[t] = sm; __syncthreads();
  for (int s = 256; s > 0; s >>= 1) { if (t < s) red[t] += red[t + s]; __syncthreads(); }
  if (t == 0) { maxb[b] = M; sumb[b] = red[0]; }
}

// ---------------- kernel 3: normalize -> flat node_weights table -------------
__global__ __launch_bounds__(256) void k_norm(const float* __restrict__ logits,
                                              const float* __restrict__ maxb,
                                              const float* __restrict__ sumb,
                                              float* __restrict__ nw) {
  const long i = (long)blockIdx.x * 256 + threadIdx.x;
  const int b = (int)(i >> 15);              // N == 2^15
  nw[i] = __expf(logits[i] - maxb[b]) / sumb[b];
}

// ---------------- kernel 4: edge softmax + gather + blend (1 wave = 1 node) --
__global__ __launch_bounds__(256) void k_edge(const float* __restrict__ el,
                                              const long long* __restrict__ nidx,
                                              const float* __restrict__ nw,
                                              float* __restrict__ out) {
  const int t = threadIdx.x;
  const int lane = t & 31;
  const long node = (long)blockIdx.x * 8 + (t >> 5);
  const long base = node * KK + lane;

  float e = el[base];
  float m = e;
#pragma unroll
  for (int off = 16; off > 0; off >>= 1) m = fmaxf(m, __shfl_xor(m, off, 32));
  float ex = __expf(e - m);
  float s = ex;
#pragma unroll
  for (int off = 16; off > 0; off >>= 1) s += __shfl_xor(s, off, 32);

  const long long idx = nidx[base];          // flat index into [B*N]
  float c = (ex / s) * nw[idx];
#pragma unroll
  for (int off = 16; off > 0; off >>= 1) c += __shfl_xor(c, off, 32);

  if (lane == 0) out[node] = 0.5f * (nw[node] + c);
}

extern "C" void kernel_launch(void* const* d_in, const int* in_sizes, int n_in,
                              void* d_out, int out_size, void* d_ws, size_t ws_size,
                              hipStream_t stream) {
  const float*     node_feats   = (const float*)d_in[0];
  const float*     edge_logits  = (const float*)d_in[1];
  const long long* neighbor_idx = (const long long*)d_in[2];   // int64 per reference
  const float*     v1 = (const float*)d_in[3];
  const float*     g1 = (const float*)d_in[4];
  const float*     b1 = (const float*)d_in[5];
  const float*     v2 = (const float*)d_in[6];
  const float*     g2 = (const float*)d_in[7];
  // d_in[8] = b2: adds a constant to every logit along the softmax axis -> no-op.

  float* ws     = (float*)d_ws;
  float* w1P    = ws;                        // 8192 f (pair-interleaved)
  float* w2n    = w1P + DD * HH;             // 64 f
  float* maxb   = w2n + HH;                  // 8 f
  float* sumb   = maxb + BB;                 // 8 f
  float* logits = sumb + BB;                 // B*N f
  float* nw     = logits + (long)BB * NN;    // B*N f   (~2.1 MB total ws)
  float* outp   = (float*)d_out;

  k_prep <<<1, HH, 0, stream>>>(v1, g1, v2, g2, w1P, w2n);
  k_mlp  <<<BB * NN / 128, 256, 0, stream>>>(node_feats, w1P, w2n, b1, logits);
  k_stats<<<BB, 512, 0, stream>>>(logits, maxb, sumb);
  k_norm <<<BB * NN / 256, 256, 0, stream>>>(logits, maxb, sumb, nw);
  k_edge <<<BB * NN / 8, 256, 0, stream>>>(edge_logits, neighbor_idx, nw, outp);
}